// BottleneckAdapter_36240934043753
// MI455X (gfx1250) — compile-verified
//
#include <hip/hip_runtime.h>
#include <math.h>

#define D_MODEL 1280
#define D_BOT   64
#define LN_EPS  1e-5f

typedef float v2f __attribute__((ext_vector_type(2)));
typedef float v8f __attribute__((ext_vector_type(8)));

__device__ __forceinline__ v2f ld2(const float* p) { return *(const v2f*)p; }

// ---------------------------------------------------------------------------
// Prep: Wg[k] = sum_d w_down[k,d]*gamma[d] ; Wb[k] = sum_d w_down[k,d]*beta[d]
// Optionally also writes Wp[k,d] = w_down[k,d]*gamma[d]  (pre-folded weights)
// ---------------------------------------------------------------------------
__global__ __launch_bounds__(256) void adapter_prep_kernel(
    const float* __restrict__ wdn, const float* __restrict__ gamma,
    const float* __restrict__ beta, float* __restrict__ wg, float* __restrict__ wb,
    float* __restrict__ wp /* may be null */)
{
    __shared__ float s1[256];
    __shared__ float s2[256];
    const int k = blockIdx.x;
    const float* row = wdn + k * D_MODEL;
    float a = 0.0f, b = 0.0f;
    for (int d = threadIdx.x; d < D_MODEL; d += 256) {
        float w  = row[d];
        float wgm = w * gamma[d];
        a += wgm;
        b += w * beta[d];
        if (wp) wp[k * D_MODEL + d] = wgm;
    }
    s1[threadIdx.x] = a; s2[threadIdx.x] = b;
    __syncthreads();
    for (int s = 128; s > 0; s >>= 1) {
        if ((int)threadIdx.x < s) {
            s1[threadIdx.x] += s1[threadIdx.x + s];
            s2[threadIdx.x] += s2[threadIdx.x + s];
        }
        __syncthreads();
    }
    if (threadIdx.x == 0) { wg[k] = s1[0]; wb[k] = s2[0]; }
}

// ---------------------------------------------------------------------------
// Main fused kernel: one wave (32 threads) per 32 tokens (two M=16 WMMA tiles)
// USE_WP=true: down-proj B operand comes pre-folded with gamma from workspace.
// ---------------------------------------------------------------------------
template <bool USE_WP>
__global__ __launch_bounds__(32) void adapter_main_kernel(
    const float* __restrict__ x,   const float* __restrict__ gamma,
    const float* __restrict__ wdn, const float* __restrict__ bdn,
    const float* __restrict__ wup, const float* __restrict__ bup,
    const float* __restrict__ wg,  const float* __restrict__ wb,
    const float* __restrict__ wp,
    float* __restrict__ out, int ntok)
{
    __shared__ float2 stats[32];                       // (mean, rstd) per token
    __shared__ __align__(16) float g_lds[2 * 16 * 64]; // GELU(down) transpose buffer

    const int lane = threadIdx.x;      // 0..31
    const int lo   = lane & 15;
    const int half = lane >> 4;        // 0: K%4 in {0,1}, 1: K%4 in {2,3}
    const int coff = 2 * half;
    const int tok0 = blockIdx.x * 32;

    // clamp row indices (safe if ntok % 32 != 0; duplicated rows write identical data)
    int trow[2];
    trow[0] = min(tok0 + lo,      ntok - 1);
    trow[1] = min(tok0 + 16 + lo, ntok - 1);
    const float* xrow0 = x + (long)trow[0] * D_MODEL;
    const float* xrow1 = x + (long)trow[1] * D_MODEL;
    const float* wrow  = (USE_WP ? wp : wdn) + lo * D_MODEL;   // rows lo, lo+16*nt

    // ---- Pass 1: fused LN-stats + down-projection over K=1280 ----
    v8f acc[2][4];
#pragma unroll
    for (int m = 0; m < 2; ++m)
#pragma unroll
        for (int nt = 0; nt < 4; ++nt) acc[m][nt] = (v8f){};

    float sum0 = 0.f, sum1 = 0.f, ssq0 = 0.f, ssq1 = 0.f;

#pragma unroll 4
    for (int j = 0; j < D_MODEL / 4; ++j) {
        const int d0 = 4 * j + coff;
        v2f a0 = ld2(xrow0 + d0);
        v2f a1 = ld2(xrow1 + d0);
        v2f bb[4];
        if (USE_WP) {
#pragma unroll
            for (int nt = 0; nt < 4; ++nt)
                bb[nt] = ld2(wrow + nt * 16 * D_MODEL + d0);
        } else {
            v2f g2 = ld2(gamma + d0);
#pragma unroll
            for (int nt = 0; nt < 4; ++nt) {
                bb[nt] = ld2(wrow + nt * 16 * D_MODEL + d0);
                bb[nt].x *= g2.x; bb[nt].y *= g2.y;
            }
        }
        sum0 += a0.x + a0.y;  ssq0 += a0.x * a0.x + a0.y * a0.y;
        sum1 += a1.x + a1.y;  ssq1 += a1.x * a1.x + a1.y * a1.y;
#pragma unroll
        for (int nt = 0; nt < 4; ++nt) {
            acc[0][nt] = __builtin_amdgcn_wmma_f32_16x16x4_f32(
                false, a0, false, bb[nt], (short)0, acc[0][nt], false, false);
            acc[1][nt] = __builtin_amdgcn_wmma_f32_16x16x4_f32(
                false, a1, false, bb[nt], (short)0, acc[1][nt], false, false);
        }
    }

    // ---- LN statistics: combine the two K-halves of each token row ----
    {
        float s, q, mean, var;
        s = sum0 + __shfl_xor(sum0, 16); q = ssq0 + __shfl_xor(ssq0, 16);
        mean = s * (1.0f / D_MODEL);
        var  = q * (1.0f / D_MODEL) - mean * mean;
        if (half == 0) stats[lo]      = make_float2(mean, rsqrtf(var + LN_EPS));
        s = sum1 + __shfl_xor(sum1, 16); q = ssq1 + __shfl_xor(ssq1, 16);
        mean = s * (1.0f / D_MODEL);
        var  = q * (1.0f / D_MODEL) - mean * mean;
        if (half == 0) stats[16 + lo] = make_float2(mean, rsqrtf(var + LN_EPS));
    }
    __syncthreads();

    // ---- Down-proj epilogue: LN correction + bias + exact GELU -> LDS ----
#pragma unroll
    for (int nt = 0; nt < 4; ++nt) {
        const int k   = nt * 16 + lo;
        const float wgk = wg[k];
        const float cst = wb[k] + bdn[k];
#pragma unroll
        for (int m = 0; m < 2; ++m) {
#pragma unroll
            for (int r = 0; r < 8; ++r) {
                float2 ms = stats[m * 16 + r + 8 * half];   // (mean, rstd)
                float v = ms.y * (acc[m][nt][r] - ms.x * wgk) + cst;
                v = 0.5f * v * (1.0f + erff(v * 0.70710678118654752f)); // exact GELU
                g_lds[m * 1024 + (r + 8 * half) * 64 + k] = v;
            }
        }
    }
    __syncthreads();

    // ---- Preload GELU output in WMMA A-layout (K=64 -> 16 steps) ----
    v2f ga[2][16];
#pragma unroll
    for (int m = 0; m < 2; ++m)
#pragma unroll
        for (int j = 0; j < 16; ++j)
            ga[m][j] = *(const v2f*)&g_lds[m * 1024 + lo * 64 + 4 * j + coff];

    // ---- Pass 2: up-projection (80 N-tiles) + residual + bias epilogue ----
    for (int nt = 0; nt < D_MODEL / 16; ++nt) {
        v8f c0 = (v8f){}, c1 = (v8f){};
#pragma unroll
        for (int j = 0; j < 16; ++j) {
            v2f b = ld2(wup + (nt * 16 + lo) * D_BOT + 4 * j + coff);
            c0 = __builtin_amdgcn_wmma_f32_16x16x4_f32(
                false, ga[0][j], false, b, (short)0, c0, false, false);
            c1 = __builtin_amdgcn_wmma_f32_16x16x4_f32(
                false, ga[1][j], false, b, (short)0, c1, false, false);
        }
        const int d  = nt * 16 + lo;
        const float bu = bup[d];
#pragma unroll
        for (int r = 0; r < 8; ++r) {
            {
                int t = min(tok0 + r + 8 * half, ntok - 1);
                long idx = (long)t * D_MODEL + d;
                out[idx] = c0[r] + bu + x[idx];
            }
            {
                int t = min(tok0 + 16 + r + 8 * half, ntok - 1);
                long idx = (long)t * D_MODEL + d;
                out[idx] = c1[r] + bu + x[idx];
            }
        }
    }
}

// ---------------------------------------------------------------------------
extern "C" void kernel_launch(void* const* d_in, const int* in_sizes, int n_in,
                              void* d_out, int out_size, void* d_ws, size_t ws_size,
                              hipStream_t stream) {
    const float* x     = (const float*)d_in[0];
    const float* gamma = (const float*)d_in[1];
    const float* beta  = (const float*)d_in[2];
    const float* wdn   = (const float*)d_in[3];
    const float* bdn   = (const float*)d_in[4];
    const float* wup   = (const float*)d_in[5];
    const float* bup   = (const float*)d_in[6];
    float* out = (float*)d_out;

    // workspace layout: wg[64] | wb[64] | (optional) Wp[64*1280]
    float* wg = (float*)d_ws;
    float* wb = wg + D_BOT;
    const size_t need_wp = (size_t)(2 * D_BOT + D_BOT * D_MODEL) * sizeof(float);
    const bool use_wp = (ws_size >= need_wp);
    float* wp = use_wp ? (wb + D_BOT) : nullptr;

    adapter_prep_kernel<<<D_BOT, 256, 0, stream>>>(wdn, gamma, beta, wg, wb, wp);

    const int ntok = in_sizes[0] / D_MODEL;        // 16*1500 = 24000
    const int nblk = (ntok + 31) / 32;             // 750
    if (use_wp) {
        adapter_main_kernel<true><<<nblk, 32, 0, stream>>>(
            x, gamma, wdn, bdn, wup, bup, wg, wb, wp, out, ntok);
    } else {
        adapter_main_kernel<false><<<nblk, 32, 0, stream>>>(
            x, gamma, wdn, bdn, wup, bup, wg, wb, wp, out, ntok);
    }
}